// GPTClassification_50680614092843
// MI455X (gfx1250) — compile-verified
//
#include <hip/hip_runtime.h>
#include <hip/hip_bf16.h>
#include <math.h>

// ---------------------------------------------------------------------------
// Adaptive-softmax LM loss for MI455X (gfx1250), fp32 WMMA path + TDM staging.
// DMODEL=1024, VOCAB=50257, cutoffs (100,1000,10000,50257), N=2046 rows.
// ---------------------------------------------------------------------------

typedef float v2f __attribute__((ext_vector_type(2)));
typedef float v8f __attribute__((ext_vector_type(8)));
typedef unsigned int v4u __attribute__((ext_vector_type(4)));
typedef int v4i __attribute__((ext_vector_type(4)));
typedef int v8i __attribute__((ext_vector_type(8)));

#define NROWS 2046
#define NPAD  2048
#define DM    1024
#define TWAVES 8      // waves per tail block
#define SPLIT2 2      // column splits, tail cluster 2
#define SPLIT3 4      // column splits, tail cluster 3
#define LSE_EMPTY (-1e30f)

__device__ __forceinline__ float wave_reduce_sum(float v) {
  #pragma unroll
  for (int off = 16; off >= 1; off >>= 1) v += __shfl_xor(v, off, 32);
  return v;
}

__device__ __forceinline__ float block_reduce_sum(float v, float* red) {
  const int wave = threadIdx.x >> 5, lane = threadIdx.x & 31;
  v = wave_reduce_sum(v);
  __syncthreads();
  if (lane == 0) red[wave] = v;
  __syncthreads();
  const int nw = blockDim.x >> 5;
  float t = (threadIdx.x < nw) ? red[threadIdx.x] : 0.f;
  if (wave == 0) {
    t = wave_reduce_sum(t);
    if (lane == 0) red[0] = t;
  }
  __syncthreads();
  return red[0];
}

__device__ __forceinline__ float logaddexp(float a, float b) {
  const float hi = fmaxf(a, b), lo = fminf(a, b);
  return hi + __logf(1.f + __expf(lo - hi));
}

// ---------------------------------------------------------------------------
// LayerNorm of shifted rows; pad rows 2046..2047 with zeros. 256 thr/block.
// ---------------------------------------------------------------------------
__global__ void ln_kernel(const float* __restrict__ x,
                          const int* __restrict__ targets,
                          const float* __restrict__ gamma,
                          const float* __restrict__ beta,
                          float* __restrict__ xs,
                          int* __restrict__ tsb) {
  __shared__ float red[8];
  const int n = blockIdx.x;
  const int tid = threadIdx.x;
  if (n >= NROWS) {
    for (int i = tid; i < DM; i += 256) xs[(size_t)n * DM + i] = 0.f;
    if (tid == 0) tsb[n] = 0;
    return;
  }
  const int b = n / 1023, s = n % 1023;
  const float* xr = x + (size_t)(b * 1024 + s) * DM;
  float vals[4], acc = 0.f;
  #pragma unroll
  for (int i = 0; i < 4; ++i) { vals[i] = xr[tid + 256 * i]; acc += vals[i]; }
  const float mu = block_reduce_sum(acc, red) * (1.f / DM);
  acc = 0.f;
  #pragma unroll
  for (int i = 0; i < 4; ++i) { float d = vals[i] - mu; acc += d * d; }
  const float var = block_reduce_sum(acc, red) * (1.f / DM);
  const float rinv = rsqrtf(var + 1e-5f);
  #pragma unroll
  for (int i = 0; i < 4; ++i) {
    const int c = tid + 256 * i;
    xs[(size_t)n * DM + c] = (vals[i] - mu) * rinv * gamma[c] + beta[c];
  }
  if (tid == 0) tsb[n] = targets[b * 1024 + s + 1];
}

// ---------------------------------------------------------------------------
// C[N x M] = A[N x K] * B[M x K]^T via V_WMMA_F32_16X16X4_F32.
// One wave per (2 row tiles) x (16*CPW cols): per K-step, 2+CPW loads feed
// 2*CPW WMMAs. Fragments prefetched 2 K-steps ahead (rotating buffer) so
// at least one load group is in flight during the WMMAs.
// grid = (ceil(M/(16*CPW)), N/32), block = 32.
// ---------------------------------------------------------------------------
template <int CPW>
__global__ void wmma_gemm(const float* __restrict__ A,
                          const float* __restrict__ B,
                          float* __restrict__ C,
                          int K, int M, int ldc) {
  const int lane = threadIdx.x & 31;
  const int half = lane >> 4;
  const int l16  = lane & 15;
  const int rowb = blockIdx.y * 32;
  const int colb = blockIdx.x * 16 * CPW;
  const float* ap0 = A + (size_t)(rowb + l16) * K + 2 * half;
  const float* ap1 = A + (size_t)(rowb + 16 + l16) * K + 2 * half;
  const float* bp[CPW];
  #pragma unroll
  for (int c = 0; c < CPW; ++c) {
    int bm = colb + 16 * c + l16;
    if (bm > M - 1) bm = M - 1;                 // clamp; consumers mask pads
    bp[c] = B + (size_t)bm * K + 2 * half;
  }
  v8f acc[2][CPW];
  #pragma unroll
  for (int r = 0; r < 2; ++r)
    #pragma unroll
    for (int c = 0; c < CPW; ++c) acc[r][c] = (v8f){};

  const int nsteps = K >> 2;
  v2f a_buf[2][2];        // [parity][row tile]
  v2f b_buf[2][CPW];      // [parity][col tile]
  #pragma unroll
  for (int p = 0; p < 2; ++p) {
    a_buf[p][0] = *(const v2f*)(ap0 + 4 * p);
    a_buf[p][1] = *(const v2f*)(ap1 + 4 * p);
    #pragma unroll
    for (int c = 0; c < CPW; ++c) b_buf[p][c] = *(const v2f*)(bp[c] + 4 * p);
  }
  for (int s = 0; s < nsteps; ++s) {
    const int p = s & 1;
    #pragma unroll
    for (int c = 0; c < CPW; ++c) {
      acc[0][c] = __builtin_amdgcn_wmma_f32_16x16x4_f32(false, a_buf[p][0], false, b_buf[p][c],
                                                        (short)0, acc[0][c], false, false);
      acc[1][c] = __builtin_amdgcn_wmma_f32_16x16x4_f32(false, a_buf[p][1], false, b_buf[p][c],
                                                        (short)0, acc[1][c], false, false);
    }
    const int s2 = s + 2;
    if (s2 < nsteps) {                           // refill this parity slot
      a_buf[p][0] = *(const v2f*)(ap0 + 4 * s2);
      a_buf[p][1] = *(const v2f*)(ap1 + 4 * s2);
      #pragma unroll
      for (int c = 0; c < CPW; ++c) b_buf[p][c] = *(const v2f*)(bp[c] + 4 * s2);
    }
  }
  #pragma unroll
  for (int rt = 0; rt < 2; ++rt)
    #pragma unroll
    for (int c = 0; c < CPW; ++c)
      #pragma unroll
      for (int r = 0; r < 8; ++r)
        C[(size_t)(rowb + 16 * rt + r + 8 * half) * ldc + colb + 16 * c + l16] = acc[rt][c][r];
}

// ---------------------------------------------------------------------------
// Stage a [16 x DI] fp32 tile (row stride DI) from global into LDS via the
// Tensor Data Mover. Must be called by wave 0 only (uniform args).
// ---------------------------------------------------------------------------
template <int DI>
__device__ __forceinline__ void tdm_load_tile(const float* gsrc, float* lds) {
#if __has_builtin(__builtin_amdgcn_tensor_load_to_lds)
  const unsigned long long ga = (unsigned long long)(uintptr_t)gsrc;
  const unsigned int ldsa = (unsigned int)(uintptr_t)lds;   // low 32b = LDS offset
  v4u g0;
  g0.x = 1u;                                   // count=1, no gather
  g0.y = ldsa;                                 // lds_addr [63:32]
  g0.z = (unsigned int)(ga & 0xFFFFFFFFull);   // global_addr [95:64]
  g0.w = (unsigned int)(ga >> 32) | 0x80000000u; // ga[120:96] | type=2 [127:126]
  v8i g1;
  g1[0] = 0x00020000;            // wg_mask=0, data_size=2 (4B), no flags
  g1[1] = (DI & 0xFFFF) << 16;   // tensor_dim0 = DI (lo16)
  g1[2] = 16 << 16;              // tensor_dim0 hi=0 | tensor_dim1 = 16 (lo16)
  g1[3] = DI << 16;              // tensor_dim1 hi=0 | tile_dim0 = DI
  g1[4] = 16;                    // tile_dim1 = 16, tile_dim2 = 0
  g1[5] = DI;                    // tensor_dim0_stride = DI elements
  g1[6] = 0;
  g1[7] = 0;
  v4i gz = (v4i){0, 0, 0, 0};
#if __clang_major__ >= 23
  v8i gz8 = (v8i){0, 0, 0, 0, 0, 0, 0, 0};
  __builtin_amdgcn_tensor_load_to_lds(g0, g1, gz, gz, gz8, 0);
#else
  __builtin_amdgcn_tensor_load_to_lds(g0, g1, gz, gz, 0);
#endif
  __builtin_amdgcn_s_wait_tensorcnt(0);
#else
  for (int i = threadIdx.x & 31; i < 16 * DI; i += 32) lds[i] = gsrc[i];
#endif
}

// ---------------------------------------------------------------------------
// Fused tail cluster: logits = P[16 x DI] @ W[M x DI]^T streamed through an
// online log-sum-exp; captures target-column logit. grid = (nsplit, 128):
// blockIdx.x selects a contiguous column-tile range (for occupancy),
// blockIdx.y the 16-row tile. Partial LSE per split; the split owning the
// target's tile writes the target logit (unique writer).
// ---------------------------------------------------------------------------
template <int DI>
__global__ void tail_lse_kernel(const float* __restrict__ P,   // [NPAD x DI]
                                const float* __restrict__ W,   // [M x DI]
                                const int* __restrict__ tsb,
                                float* __restrict__ lse_out,   // [nsplit x NPAD]
                                float* __restrict__ tg_out,    // [NPAD]
                                int M, int lo, int nsplit) {
  __shared__ float sa[16 * DI];
  __shared__ int   stgt[16];
  __shared__ float stargv[16];
  __shared__ float sm[TWAVES][16], ss[TWAVES][16];
  const int tid  = threadIdx.x;      // 0 .. 32*TWAVES-1
  const int wave = tid >> 5;
  const int lane = tid & 31;
  const int half = lane >> 4;
  const int l16  = lane & 15;
  const int split = blockIdx.x;
  const int rowb = blockIdx.y * 16;
  constexpr int KF = DI / 4;         // K fragments per tile

  const int ntiles = (M + 15) >> 4;
  const int tilesPer = (ntiles + nsplit - 1) / nsplit;
  const int t0 = split * tilesPer;
  const int tEnd = min(t0 + tilesPer, ntiles);

  if (wave == 0)                      // TDM DMA of the proj tile into LDS
    tdm_load_tile<DI>(P + (size_t)rowb * DI, sa);
  if (tid < 16) {
    int t = tsb[rowb + tid] - lo;     // clip like the reference
    t = t < 0 ? 0 : (t > M - 1 ? M - 1 : t);
    stgt[tid] = t;
    stargv[tid] = 0.f;
  }
  __syncthreads();

  float m8[8], s8[8];
  #pragma unroll
  for (int r = 0; r < 8; ++r) { m8[r] = -__builtin_inff(); s8[r] = 0.f; }

  const float* apl = sa + l16 * DI + 2 * half;

  if (DI <= 64) {
    // ---- pipelined path: A in registers, B double-buffered across tiles ----
    v2f af[KF];
    #pragma unroll
    for (int j = 0; j < KF; ++j) af[j] = *(const v2f*)(apl + 4 * j);

    int ct = t0 + wave;
    if (ct < tEnd) {
      int mcol = ct * 16 + l16;
      int bm = mcol > M - 1 ? M - 1 : mcol;
      v2f bcur[KF];
      #pragma unroll
      for (int j = 0; j < KF; ++j)
        bcur[j] = *(const v2f*)(W + (size_t)bm * DI + 2 * half + 4 * j);
      while (true) {
        const int ctn = ct + TWAVES;
        v2f bnxt[KF];
        if (ctn < tEnd) {                        // prefetch next tile's B
          int mcoln = ctn * 16 + l16;
          int bmn = mcoln > M - 1 ? M - 1 : mcoln;
          #pragma unroll
          for (int j = 0; j < KF; ++j)
            bnxt[j] = *(const v2f*)(W + (size_t)bmn * DI + 2 * half + 4 * j);
        }
        v8f acc = {};
        #pragma unroll
        for (int j = 0; j < KF; ++j)
          acc = __builtin_amdgcn_wmma_f32_16x16x4_f32(false, af[j], false, bcur[j],
                                                      (short)0, acc, false, false);
        if (mcol < M) {
          #pragma unroll
          for (int r = 0; r < 8; ++r) {
            const float v = acc[r];
            if (v > m8[r]) { s8[r] = s8[r] * __expf(m8[r] - v) + 1.f; m8[r] = v; }
            else           { s8[r] += __expf(v - m8[r]); }
            const int row = r + 8 * half;
            if (stgt[row] == mcol) stargv[row] = v;
          }
        }
        if (ctn >= tEnd) break;
        ct = ctn;
        mcol = ct * 16 + l16;
        #pragma unroll
        for (int j = 0; j < KF; ++j) bcur[j] = bnxt[j];
      }
    }
  } else {
    // ---- direct path (DI=256: only ~57 tiles) ----
    for (int ct = t0 + wave; ct < tEnd; ct += TWAVES) {
      const int mcol = ct * 16 + l16;
      const int bm = mcol > M - 1 ? M - 1 : mcol;
      const float* bp = W + (size_t)bm * DI + 2 * half;
      v8f acc = {};
      #pragma unroll
      for (int k = 0; k < DI; k += 4) {
        v2f a = *(const v2f*)(apl + k);
        v2f b = *(const v2f*)(bp + k);
        acc = __builtin_amdgcn_wmma_f32_16x16x4_f32(false, a, false, b,
                                                    (short)0, acc, false, false);
      }
      if (mcol < M) {
        #pragma unroll
        for (int r = 0; r < 8; ++r) {
          const float v = acc[r];
          if (v > m8[r]) { s8[r] = s8[r] * __expf(m8[r] - v) + 1.f; m8[r] = v; }
          else           { s8[r] += __expf(v - m8[r]); }
          const int row = r + 8 * half;
          if (stgt[row] == mcol) stargv[row] = v;
        }
      }
    }
  }

  // combine the 16 lanes (columns) that share each row group
  #pragma unroll
  for (int off = 8; off >= 1; off >>= 1) {
    #pragma unroll
    for (int r = 0; r < 8; ++r) {
      const float m2 = __shfl_xor(m8[r], off, 32);
      const float s2 = __shfl_xor(s8[r], off, 32);
      if (s2 > 0.f) {
        if (m2 > m8[r]) { s8[r] = s8[r] * __expf(m8[r] - m2) + s2; m8[r] = m2; }
        else            { s8[r] += s2 * __expf(m2 - m8[r]); }
      }
    }
  }
  if (l16 == 0) {
    #pragma unroll
    for (int r = 0; r < 8; ++r) {
      sm[wave][r + 8 * half] = m8[r];
      ss[wave][r + 8 * half] = s8[r];
    }
  }
  __syncthreads();
  if (tid < 16) {
    float m = sm[0][tid], s = ss[0][tid];
    #pragma unroll
    for (int w = 1; w < TWAVES; ++w) {
      const float m2 = sm[w][tid], s2 = ss[w][tid];
      if (s2 > 0.f) {
        if (m2 > m) { s = s * __expf(m - m2) + s2; m = m2; }
        else        { s += s2 * __expf(m2 - m); }
      }
    }
    lse_out[(size_t)split * NPAD + rowb + tid] = (s > 0.f) ? (m + __logf(s)) : LSE_EMPTY;
    const int ownerSplit = (stgt[tid] >> 4) / tilesPer;   // split owning target tile
    if (ownerSplit == split) tg_out[rowb + tid] = stargv[tid];
  }
}

// ---------------------------------------------------------------------------
// Head LSE + split-merge + cluster selection + mean. Single block.
// ---------------------------------------------------------------------------
__global__ void final_kernel(const float* __restrict__ hl,    // [NPAD x 112]
                             const float* __restrict__ lse1, const float* __restrict__ tg1,
                             const float* __restrict__ lse2, const float* __restrict__ tg2,
                             const float* __restrict__ lse3, const float* __restrict__ tg3,
                             const int* __restrict__ tsb,
                             float* __restrict__ out) {
  __shared__ float red[32];
  float local = 0.f;
  for (int n = threadIdx.x; n < NROWS; n += blockDim.x) {
    const float* h = hl + (size_t)n * 112;
    float m = -__builtin_inff();
    for (int j = 0; j < 103; ++j) m = fmaxf(m, h[j]);
    float s = 0.f;
    for (int j = 0; j < 103; ++j) s += __expf(h[j] - m);
    const float lseh = m + __logf(s);
    const int t = tsb[n];
    const int th = t > 99 ? 99 : (t < 0 ? 0 : t);
    float lp = h[th] - lseh;
    if (t >= 100 && t < 1000) {
      lp = (h[100] - lseh) + (tg1[n] - lse1[n]);
    } else if (t >= 1000 && t < 10000) {
      float l = lse2[n];
      #pragma unroll
      for (int sp = 1; sp < SPLIT2; ++sp) l = logaddexp(l, lse2[(size_t)sp * NPAD + n]);
      lp = (h[101] - lseh) + (tg2[n] - l);
    } else if (t >= 10000) {
      float l = lse3[n];
      #pragma unroll
      for (int sp = 1; sp < SPLIT3; ++sp) l = logaddexp(l, lse3[(size_t)sp * NPAD + n]);
      lp = (h[102] - lseh) + (tg3[n] - l);
    }
    local += lp;
  }
  const float total = block_reduce_sum(local, red);
  if (threadIdx.x == 0) out[0] = -total / (float)NROWS;
}

// ---------------------------------------------------------------------------
extern "C" void kernel_launch(void* const* d_in, const int* in_sizes, int n_in,
                              void* d_out, int out_size, void* d_ws, size_t ws_size,
                              hipStream_t stream) {
  const float* x       = (const float*)d_in[0];
  const int*   targets = (const int*)  d_in[1];
  const float* gamma   = (const float*)d_in[2];
  const float* beta    = (const float*)d_in[3];
  const float* head_w  = (const float*)d_in[4];
  const float* t1_p1   = (const float*)d_in[5];
  const float* t1_p2   = (const float*)d_in[6];
  const float* t2_p1   = (const float*)d_in[7];
  const float* t2_p2   = (const float*)d_in[8];
  const float* t3_p1   = (const float*)d_in[9];
  const float* t3_p2   = (const float*)d_in[10];

  float* ws   = (float*)d_ws;
  float* xs   = ws;                        // 2048*1024
  float* p1   = xs  + (size_t)NPAD * DM;   // 2048*256
  float* p2   = p1  + (size_t)NPAD * 256;  // 2048*64
  float* p3   = p2  + (size_t)NPAD * 64;   // 2048*16
  float* hl   = p3  + (size_t)NPAD * 16;   // 2048*112
  float* lse1 = hl  + (size_t)NPAD * 112;  // 1 split
  float* tg1  = lse1 + (size_t)NPAD * 1;
  float* lse2 = tg1  + NPAD;               // SPLIT2 splits
  float* tg2  = lse2 + (size_t)NPAD * SPLIT2;
  float* lse3 = tg2  + NPAD;               // SPLIT3 splits
  float* tg3  = lse3 + (size_t)NPAD * SPLIT3;
  int*   tsb  = (int*)(tg3 + NPAD);

  ln_kernel<<<NPAD, 256, 0, stream>>>(x, targets, gamma, beta, xs, tsb);

  // projections + head logits (C = A . B^T); 2 row tiles per wave
  wmma_gemm<2><<<dim3(8, NPAD / 32), 32, 0, stream>>>(xs, t1_p1, p1, DM, 256, 256);
  wmma_gemm<2><<<dim3(2, NPAD / 32), 32, 0, stream>>>(xs, t2_p1, p2, DM,  64,  64);
  wmma_gemm<1><<<dim3(1, NPAD / 32), 32, 0, stream>>>(xs, t3_p1, p3, DM,  16,  16);
  wmma_gemm<1><<<dim3(7, NPAD / 32), 32, 0, stream>>>(xs, head_w, hl, DM, 103, 112);

  // fused second-matmul + log-sum-exp per tail cluster (column-split blocks)
  tail_lse_kernel<256><<<dim3(1,      NPAD / 16), 32 * TWAVES, 0, stream>>>(p1, t1_p2, tsb, lse1, tg1,   900,   100, 1);
  tail_lse_kernel< 64><<<dim3(SPLIT2, NPAD / 16), 32 * TWAVES, 0, stream>>>(p2, t2_p2, tsb, lse2, tg2,  9000,  1000, SPLIT2);
  tail_lse_kernel< 16><<<dim3(SPLIT3, NPAD / 16), 32 * TWAVES, 0, stream>>>(p3, t3_p2, tsb, lse3, tg3, 40257, 10000, SPLIT3);

  final_kernel<<<1, 1024, 0, stream>>>(hl, lse1, tg1, lse2, tg2, lse3, tg3, tsb,
                                       (float*)d_out);
  (void)in_sizes; (void)n_in; (void)out_size; (void)ws_size;
}